// DUSPSA_56075093016838
// MI455X (gfx1250) — compile-verified
//
#include <hip/hip_runtime.h>
#include <math.h>

// ---------------------------------------------------------------------------
// SPSA on MI455X (gfx1250).
//
// 16384 independent 2-param problems x 1000 sequential steps.
// Memory floor: delta_bits = 128 MB -> ~5.5us @ 23.3 TB/s.
// Compute: ~0.33 GFLOP total -> irrelevant; no matmul structure -> no WMMA.
// Bottleneck: per-step dependent VALU chain + per-step 8B/thread load with
// only 512 waves of parallelism. Strategy:
//   * 2 elements/thread -> 2 independent chains of ILP per lane.
//   * Stream delta_bits with CDNA5 async copies (global_load_async_to_lds_b128,
//     ASYNCcnt) quad-buffered 3 chunks (24 steps) ahead; consume via ds_load.
//   * Per-step scalars ck/ak/1/(2ck) precomputed once into d_ws.
// ---------------------------------------------------------------------------

#define ALPHA_C 0.602f
#define GAMMA_C 0.101f
#define QCOEF   8.0f

#define CH          8                       // steps per chunk (1000 % 8 == 0)
#define NBUF        4                       // LDS buffers per wave (quad buffer)
#define STEP_BYTES  512                     // 64 elems * 8B per wave per step
#define CHUNK_BYTES (CH * STEP_BYTES)       // 4096
#define WAVE_LDS    (NBUF * CHUNK_BYTES)    // 16384
#define BLOCK_LDS   (8 * WAVE_LDS)          // 131072 (8 waves) <= 320KB WGP

// --------------------------- per-step scalars ------------------------------
__global__ void spsa_scalars(const float* __restrict__ a,
                             const float* __restrict__ c,
                             float* __restrict__ ck,
                             float* __restrict__ ak,
                             float* __restrict__ i2c,
                             int n, float A)
{
    int i = blockIdx.x * blockDim.x + threadIdx.x;
    if (i >= n) return;
    float kf  = (float)(i + 1);                         // k = 1..n
    float ckv = c[i] / powf(kf + 1.0f, GAMMA_C);        // c/(k+1)^gamma
    float akv = a[i] / powf(kf + 1.0f + A, ALPHA_C);    // a/(k+1+A)^alpha
    ck[i]  = ckv;
    ak[i]  = akv;
    i2c[i] = 1.0f / (2.0f * ckv);
}

// ------------------------------ step math ----------------------------------
// Mirrors reference: ghat_p = (f(xp)-f(xm)) / (2*ck*d_p); 1/d = d for d=+-1.
__device__ __forceinline__ void spsa_step(float& x0, float& x1,
                                          float d0, float d1,
                                          float ckk, float akk, float ii)
{
    float xp0 = fmaf( ckk, d0, x0), xm0 = fmaf(-ckk, d0, x0);
    float xp1 = fmaf( ckk, d1, x1), xm1 = fmaf(-ckk, d1, x1);
    float fp  = fmaf(QCOEF, xp1 * xp1, xp0 * xp0);
    float fm  = fmaf(QCOEF, xm1 * xm1, xm0 * xm0);
    float u   = akk * ((fp - fm) * ii);   // ak * (f(xp)-f(xm)) / (2*ck)
    x0 = fmaf(-u, d0, x0);                // x - u*d  (== x - ak*ghat)
    x1 = fmaf(-u, d1, x1);
}

// ------------------------------ main kernel --------------------------------
__global__ void __launch_bounds__(256, 1)
spsa_main(const float* __restrict__ X0,
          const int*   __restrict__ delta,
          const float* __restrict__ ck,
          const float* __restrict__ ak,
          const float* __restrict__ i2c,
          float*       __restrict__ out,
          int bs, int nchunks)
{
    extern __shared__ unsigned char smem[];            // BLOCK_LDS bytes dynamic

    const int lane = threadIdx.x & 31;
    const int w    = threadIdx.x >> 5;                 // wave id in block
    const int elemBase = blockIdx.x * 512 + w * 64;    // wave owns 64 elements
    const unsigned waveLds = (unsigned)(w * WAVE_LDS);

    // two elements per thread (ILP on the dependent chain)
    const int e0 = elemBase + lane;
    const int e1 = e0 + 32;
    float x00 = fmaf(X0[2 * e0 + 0], 20.f, -10.f);
    float x01 = fmaf(X0[2 * e0 + 1], 20.f, -10.f);
    float x10 = fmaf(X0[2 * e1 + 0], 20.f, -10.f);
    float x11 = fmaf(X0[2 * e1 + 1], 20.f, -10.f);

    const unsigned long long stepStride = (unsigned long long)bs * 8ull;
    const unsigned long long gwave =
        (unsigned long long)(size_t)delta +
        (unsigned long long)elemBase * 8ull +
        (unsigned long long)(lane * 16);               // lane carries 16B/step

    // Issue one chunk: CH async b128 copies (32 lanes x 16B = 512B = 1 step).
    auto issue_chunk = [&](int c) {
        unsigned ldsb = waveLds + (unsigned)((c & (NBUF - 1)) * CHUNK_BYTES)
                                + (unsigned)(lane * 16);
        unsigned long long g = gwave + (unsigned long long)(c * CH) * stepStride;
#pragma unroll
        for (int s = 0; s < CH; ++s) {
            asm volatile("global_load_async_to_lds_b128 %0, %1, off"
                         :: "v"(ldsb), "v"(g) : "memory");
            ldsb += STEP_BYTES;
            g    += stepStride;
        }
    };

    // Prime the pipeline: up to 3 chunks (24 steps) in flight.
    const int prefetch = nchunks < 3 ? nchunks : 3;
    for (int c = 0; c < prefetch; ++c) issue_chunk(c);

    for (int c = 0; c < nchunks; ++c) {
        // Async copies complete in order within a wave: leaving <= 2 newer
        // chunks (16 instr) outstanding guarantees chunk c fully landed.
        const int newer = nchunks - 1 - c;
        if (newer >= 2)      asm volatile("s_wait_asynccnt 16" : : : "memory");
        else if (newer == 1) asm volatile("s_wait_asynccnt 8"  : : : "memory");
        else                 asm volatile("s_wait_asynccnt 0"  : : : "memory");

        // Uniform per-step scalars for this chunk (hoisted scalar loads).
        float cks[CH], aks[CH], i2s[CH];
#pragma unroll
        for (int s = 0; s < CH; ++s) {
            const int k = c * CH + s;
            cks[s] = ck[k]; aks[s] = ak[k]; i2s[s] = i2c[k];
        }

        const unsigned bufb = waveLds + (unsigned)((c & (NBUF - 1)) * CHUNK_BYTES);
#pragma unroll
        for (int s = 0; s < CH; ++s) {
            const unsigned off0 = bufb + (unsigned)(s * STEP_BYTES)
                                       + (unsigned)(lane * 8);
            const long long v0 = *(const long long*)(smem + off0);        // e0
            const long long v1 = *(const long long*)(smem + off0 + 256);  // e1
            const float d00 = ((int)v0)         ? 1.f : -1.f;   // 2*bit-1
            const float d01 = ((int)(v0 >> 32)) ? 1.f : -1.f;
            const float d10 = ((int)v1)         ? 1.f : -1.f;
            const float d11 = ((int)(v1 >> 32)) ? 1.f : -1.f;
            spsa_step(x00, x01, d00, d01, cks[s], aks[s], i2s[s]);
            spsa_step(x10, x11, d10, d11, cks[s], aks[s], i2s[s]);
        }

        // Refill the buffer we will next recycle (WAR: reads must retire).
        const int nc = c + 3;
        if (nc < nchunks) {
            asm volatile("s_wait_dscnt 0" : : : "memory");
            issue_chunk(nc);
        }
    }

    out[2 * e0 + 0] = x00;  out[2 * e0 + 1] = x01;
    out[2 * e1 + 0] = x10;  out[2 * e1 + 1] = x11;
}

// ----------------------- generic fallback (odd sizes) ----------------------
__global__ void spsa_simple(const float* __restrict__ X0,
                            const int*   __restrict__ delta,
                            const float* __restrict__ ck,
                            const float* __restrict__ ak,
                            const float* __restrict__ i2c,
                            float*       __restrict__ out,
                            int bs, int n)
{
    int e = blockIdx.x * blockDim.x + threadIdx.x;
    if (e >= bs) return;
    float x0 = fmaf(X0[2 * e + 0], 20.f, -10.f);
    float x1 = fmaf(X0[2 * e + 1], 20.f, -10.f);
    for (int k = 0; k < n; ++k) {
        const int b0 = delta[(k * bs + e) * 2 + 0];
        const int b1 = delta[(k * bs + e) * 2 + 1];
        const float d0 = b0 ? 1.f : -1.f;
        const float d1 = b1 ? 1.f : -1.f;
        spsa_step(x0, x1, d0, d1, ck[k], ak[k], i2c[k]);
    }
    out[2 * e + 0] = x0;
    out[2 * e + 1] = x1;
}

// ------------------------------- launcher ----------------------------------
extern "C" void kernel_launch(void* const* d_in, const int* in_sizes, int n_in,
                              void* d_out, int out_size, void* d_ws, size_t ws_size,
                              hipStream_t stream)
{
    const float* X0    = (const float*)d_in[0];
    const float* a     = (const float*)d_in[1];
    const float* c     = (const float*)d_in[2];
    const int*   delta = (const int*)d_in[3];
    // d_in[4] = num_itr device scalar; derive sizes host-side from in_sizes.

    const int bs = in_sizes[0] / 2;      // 16384
    const int n  = in_sizes[1];          // 1000
    const float A = floorf(0.1f * (float)n);

    float* ckp = (float*)d_ws;
    float* akp = ckp + n;
    float* i2p = akp + n;

    spsa_scalars<<<(n + 255) / 256, 256, 0, stream>>>(a, c, ckp, akp, i2p, n, A);

    if ((bs % 512) == 0 && (n % CH) == 0) {
        spsa_main<<<bs / 512, 256, BLOCK_LDS, stream>>>(
            X0, delta, ckp, akp, i2p, (float*)d_out, bs, n / CH);
    } else {
        spsa_simple<<<(bs + 255) / 256, 256, 0, stream>>>(
            X0, delta, ckp, akp, i2p, (float*)d_out, bs, n);
    }
}